// SAGCNMean_35759897706678
// MI455X (gfx1250) — compile-verified
//
#include <hip/hip_runtime.h>
#include <hip/hip_bf16.h>

// ---------------------------------------------------------------------------
// SAGCN forward for MI455X (gfx1250, wave32, WMMA).
// Shapes: B=32, N=2048, D=320 -> 160 -> 80 -> (graphconv) 40 -> 5 -> mean -> 2
// All heavy matmuls use V_WMMA_F32_16X16X4_F32 (fp32 precision = reference).
// Graph convs are fused flash-attention style: adjacency (32x2048x2048) is
// never materialized; scores -d2 <= 0 so exp needs no max-subtraction.
// ---------------------------------------------------------------------------

#define Bb   32
#define Nn   2048
#define Dd   320
#define H1d  160
#define H2d  80

typedef __attribute__((ext_vector_type(2))) float v2f;
typedef __attribute__((ext_vector_type(8))) float v8f;

__device__ __forceinline__ v8f vzero8() {
  v8f z = {0.f, 0.f, 0.f, 0.f, 0.f, 0.f, 0.f, 0.f};
  return z;
}

// D = A(16x4) * B(4x16) + C(16x16), all fp32.
__device__ __forceinline__ v8f wmma_f32(v2f a, v2f b, v8f c) {
  return __builtin_amdgcn_wmma_f32_16x16x4_f32(
      /*neg_a=*/false, a, /*neg_b=*/false, b,
      /*c_mod=*/(short)0, c, /*reuse_a=*/false, /*reuse_b=*/false);
}

// ---------------------------------------------------------------------------
// Kernel 1: h2 = relu(relu(x@W1+b1)@W2+b2).  One 16-row tile per wave,
// 4 waves / block -> 64 rows per block.  h1 tile staged per-wave in LDS.
// ---------------------------------------------------------------------------
__global__ __launch_bounds__(128) void mlp_kernel(
    const float* __restrict__ x,
    const float* __restrict__ W1, const float* __restrict__ b1,
    const float* __restrict__ W2, const float* __restrict__ b2,
    float* __restrict__ h2out) {
  __shared__ float lds_h1[4][16 * H1d];

  const int tid   = threadIdx.x;
  const int wave  = tid >> 5;
  const int lane  = tid & 31;
  const int m     = lane & 15;           // A row / B col / C col
  const int koff  = (lane >> 4) * 2;     // K sub-offset within 4-wide step
  const int mbase = (lane >> 4) * 8;     // C/D row base
  const size_t r0 = (size_t)blockIdx.x * 64 + (size_t)wave * 16;

  // ---- stage 1: 16x160 = (16x320) @ (320x160) ----
  v8f acc1[10];
#pragma unroll
  for (int t = 0; t < 10; ++t) acc1[t] = vzero8();

  const float* xrow = x + (r0 + m) * Dd;
  for (int ks = 0; ks < Dd / 4; ++ks) {
    const int k = ks * 4 + koff;
    v2f a;
    a.x = xrow[k];
    a.y = xrow[k + 1];
#pragma unroll
    for (int t = 0; t < 10; ++t) {
      v2f bb;
      bb.x = W1[(size_t)k * H1d + t * 16 + m];
      bb.y = W1[(size_t)(k + 1) * H1d + t * 16 + m];
      acc1[t] = wmma_f32(a, bb, acc1[t]);
    }
  }
  // bias + relu -> per-wave LDS (row-major 16 x 160)
#pragma unroll
  for (int t = 0; t < 10; ++t) {
    const float bias = b1[t * 16 + m];
#pragma unroll
    for (int v = 0; v < 8; ++v) {
      lds_h1[wave][(mbase + v) * H1d + t * 16 + m] =
          fmaxf(acc1[t][v] + bias, 0.f);
    }
  }

  // ---- stage 2: 16x80 = (16x160) @ (160x80) ----
  v8f acc2[5];
#pragma unroll
  for (int t = 0; t < 5; ++t) acc2[t] = vzero8();

  for (int ks = 0; ks < H1d / 4; ++ks) {
    const int k = ks * 4 + koff;
    v2f a;
    a.x = lds_h1[wave][m * H1d + k];
    a.y = lds_h1[wave][m * H1d + k + 1];
#pragma unroll
    for (int t = 0; t < 5; ++t) {
      v2f bb;
      bb.x = W2[(size_t)k * H2d + t * 16 + m];
      bb.y = W2[(size_t)(k + 1) * H2d + t * 16 + m];
      acc2[t] = wmma_f32(a, bb, acc2[t]);
    }
  }
#pragma unroll
  for (int t = 0; t < 5; ++t) {
    const float bias = b2[t * 16 + m];
#pragma unroll
    for (int v = 0; v < 8; ++v) {
      h2out[(r0 + mbase + v) * H2d + t * 16 + m] =
          fmaxf(acc2[t][v] + bias, 0.f);
    }
  }
}

// ---------------------------------------------------------------------------
// Kernel 2/4: per-node squared norms (used by both graph convs).
// ---------------------------------------------------------------------------
template <int C>
__global__ __launch_bounds__(256) void sqnorm_kernel(
    const float* __restrict__ h, float* __restrict__ sq) {
  const int i = blockIdx.x * blockDim.x + threadIdx.x;  // node in [0, B*N)
  const float* p = h + (size_t)i * C;
  float s = 0.f;
#pragma unroll 8
  for (int c = 0; c < C; ++c) {
    const float v = p[c];
    s += v * v;
  }
  sq[i] = s;
}

// ---------------------------------------------------------------------------
// Kernel 3/5: fused graph conv.
//   A[n,j] = softmax_j(-|h_n - h_j|^2);  agg = A @ h;  out = relu(agg@Wg+bg)
// Flash-style: p = exp(2*(q.k) - |q|^2 - |k|^2) <= 1 (no max needed),
// accumulate P@V and row sums in one streaming pass over key tiles.
// Block: 64 query rows (16 per wave).  Key tiles shared via LDS.
// ---------------------------------------------------------------------------
template <int C, int COUT>
__global__ __launch_bounds__(128) void graphconv_kernel(
    const float* __restrict__ h, const float* __restrict__ sqn,
    const float* __restrict__ Wg, const float* __restrict__ bg,
    float* __restrict__ out) {
  constexpr int NT = (C + 15) / 16;  // output-channel tiles
  __shared__ float lds_q[64 * C];    // query rows, later reused for agg
  __shared__ float lds_k[16 * C];    // current key tile (also V)
  __shared__ float lds_p[4 * 256];   // per-wave 16x16 probability tile

  const int tid   = threadIdx.x;
  const int wave  = tid >> 5;
  const int lane  = tid & 31;
  const int m     = lane & 15;
  const int koff  = (lane >> 4) * 2;
  const int mbase = (lane >> 4) * 8;
  const int b     = blockIdx.y;
  const int q0    = blockIdx.x * 64;
  const float* hb  = h + (size_t)b * Nn * C;
  const float* sqb = sqn + (size_t)b * Nn;

  // stage the 64 query rows (contiguous) into LDS
  for (int i = tid; i < 64 * C; i += 128) lds_q[i] = hb[(size_t)q0 * C + i];
  __syncthreads();

  const int qrow = wave * 16;
  // preload per-lane Q A-fragments (C/4 k-steps)
  v2f aq[C / 4];
#pragma unroll
  for (int ks = 0; ks < C / 4; ++ks) {
    aq[ks].x = lds_q[(qrow + m) * C + ks * 4 + koff];
    aq[ks].y = lds_q[(qrow + m) * C + ks * 4 + koff + 1];
  }
  float qs[8];
#pragma unroll
  for (int v = 0; v < 8; ++v) qs[v] = sqb[q0 + qrow + mbase + v];

  v8f acc[NT];
#pragma unroll
  for (int t = 0; t < NT; ++t) acc[t] = vzero8();
  float rs[8] = {0.f, 0.f, 0.f, 0.f, 0.f, 0.f, 0.f, 0.f};
  float* pw = lds_p + wave * 256;

  for (int j0 = 0; j0 < Nn; j0 += 16) {
    __syncthreads();  // all waves done with previous key tile
    for (int i = tid; i < 16 * C; i += 128)
      lds_k[i] = hb[(size_t)j0 * C + i];
    __syncthreads();

    const float ksq = sqb[j0 + m];  // this lane's key-column sqnorm

    // S = Q @ K^T   (16x16, over C in steps of 4)
    v8f s = vzero8();
#pragma unroll
    for (int ks = 0; ks < C / 4; ++ks) {
      v2f bk;
      bk.x = lds_k[m * C + ks * 4 + koff];
      bk.y = lds_k[m * C + ks * 4 + koff + 1];
      s = wmma_f32(aq[ks], bk, s);
    }
    // p = exp(-d2) = exp(2*s - |q|^2 - |k|^2)  (<= 1, stable)
#pragma unroll
    for (int v = 0; v < 8; ++v) {
      const float p = __expf(2.f * s[v] - qs[v] - ksq);
      rs[v] += p;
      pw[(mbase + v) * 16 + m] = p;  // stash for A-fragment relayout
    }
    // acc += P(16x16) @ V(16xC)
#pragma unroll
    for (int t = 0; t < NT; ++t) {
      const int ch = t * 16 + m;
#pragma unroll
      for (int jj = 0; jj < 4; ++jj) {
        v2f ap;
        ap.x = pw[m * 16 + jj * 4 + koff];
        ap.y = pw[m * 16 + jj * 4 + koff + 1];
        v2f bv;
        bv.x = (ch < C) ? lds_k[(jj * 4 + koff) * C + ch] : 0.f;
        bv.y = (ch < C) ? lds_k[(jj * 4 + koff + 1) * C + ch] : 0.f;
        acc[t] = wmma_f32(ap, bv, acc[t]);
      }
    }
  }

  // row sums: reduce over the 16 lanes of each half (cols 0..15)
#pragma unroll
  for (int v = 0; v < 8; ++v) {
    float r = rs[v];
    r += __shfl_xor(r, 1);
    r += __shfl_xor(r, 2);
    r += __shfl_xor(r, 4);
    r += __shfl_xor(r, 8);
    rs[v] = r;
  }

  // normalized aggregation -> reuse this wave's own rows of lds_q
#pragma unroll
  for (int t = 0; t < NT; ++t) {
    const int ch = t * 16 + m;
    if (ch < C) {
#pragma unroll
      for (int v = 0; v < 8; ++v)
        lds_q[(qrow + mbase + v) * C + ch] = acc[t][v] / rs[v];
    }
  }

  // tiny projection: out = relu(agg @ Wg + bg), 16 rows x COUT per wave
  for (int i = lane; i < 16 * COUT; i += 32) {
    const int row = i / COUT;
    const int o   = i % COUT;
    float sacc = bg[o];
    for (int c = 0; c < C; ++c)
      sacc += lds_q[(qrow + row) * C + c] * Wg[c * COUT + o];
    out[((size_t)b * Nn + q0 + qrow + row) * COUT + o] = fmaxf(sacc, 0.f);
  }
}

// ---------------------------------------------------------------------------
// Kernel 6: mean over nodes, 5->2 classifier, softmax.  One block per batch.
// ---------------------------------------------------------------------------
__global__ __launch_bounds__(256) void head_kernel(
    const float* __restrict__ h4, const float* __restrict__ Wf,
    const float* __restrict__ bf, float* __restrict__ out) {
  __shared__ float red[256 * 5];
  const int b = blockIdx.x, tid = threadIdx.x;
  float part[5] = {0.f, 0.f, 0.f, 0.f, 0.f};
  const float* hb = h4 + (size_t)b * Nn * 5;
  for (int n = tid; n < Nn; n += 256) {
#pragma unroll
    for (int c = 0; c < 5; ++c) part[c] += hb[n * 5 + c];
  }
#pragma unroll
  for (int c = 0; c < 5; ++c) red[tid * 5 + c] = part[c];
  __syncthreads();
  for (int sft = 128; sft > 0; sft >>= 1) {
    if (tid < sft) {
#pragma unroll
      for (int c = 0; c < 5; ++c) red[tid * 5 + c] += red[(tid + sft) * 5 + c];
    }
    __syncthreads();
  }
  if (tid == 0) {
    float mean[5];
#pragma unroll
    for (int c = 0; c < 5; ++c) mean[c] = red[c] * (1.f / (float)Nn);
    float l0 = bf[0], l1 = bf[1];
#pragma unroll
    for (int c = 0; c < 5; ++c) {
      l0 += mean[c] * Wf[c * 2 + 0];
      l1 += mean[c] * Wf[c * 2 + 1];
    }
    const float mx = fmaxf(l0, l1);
    const float e0 = __expf(l0 - mx), e1 = __expf(l1 - mx);
    const float inv = 1.f / (e0 + e1);
    out[b * 2 + 0] = e0 * inv;
    out[b * 2 + 1] = e1 * inv;
  }
}

// ---------------------------------------------------------------------------
extern "C" void kernel_launch(void* const* d_in, const int* in_sizes, int n_in,
                              void* d_out, int out_size, void* d_ws,
                              size_t ws_size, hipStream_t stream) {
  const float* x   = (const float*)d_in[0];
  const float* W1  = (const float*)d_in[1];
  const float* b1  = (const float*)d_in[2];
  const float* W2  = (const float*)d_in[3];
  const float* b2  = (const float*)d_in[4];
  const float* Wg1 = (const float*)d_in[5];
  const float* bg1 = (const float*)d_in[6];
  const float* Wg2 = (const float*)d_in[7];
  const float* bg2 = (const float*)d_in[8];
  const float* Wf  = (const float*)d_in[9];
  const float* bf  = (const float*)d_in[10];

  float* h2 = (float*)d_ws;                       // [32,2048,80]
  float* h3 = h2 + (size_t)Bb * Nn * H2d;         // [32,2048,40]
  float* h4 = h3 + (size_t)Bb * Nn * 40;          // [32,2048,5]
  float* sq = h4 + (size_t)Bb * Nn * 5;           // [32*2048]
  float* o  = (float*)d_out;                      // [32,2]

  mlp_kernel<<<dim3((Bb * Nn) / 64), dim3(128), 0, stream>>>(x, W1, b1, W2, b2, h2);
  sqnorm_kernel<80><<<dim3((Bb * Nn) / 256), dim3(256), 0, stream>>>(h2, sq);
  graphconv_kernel<80, 40><<<dim3(Nn / 64, Bb), dim3(128), 0, stream>>>(h2, sq, Wg1, bg1, h3);
  sqnorm_kernel<40><<<dim3((Bb * Nn) / 256), dim3(256), 0, stream>>>(h3, sq);
  graphconv_kernel<40, 5><<<dim3(Nn / 64, Bb), dim3(128), 0, stream>>>(h3, sq, Wg2, bg2, h4);
  head_kernel<<<dim3(Bb), dim3(256), 0, stream>>>(h4, Wf, bf, o);
}